// LoRALayer_41918880809105
// MI455X (gfx1250) — compile-verified
//
#include <hip/hip_runtime.h>
#include <hip/hip_bf16.h>

#define IN_DIM 256
#define OUT_DIM 64
#define RANK 3
#define WB_STRIDE 260  // stride % 64 == 4 -> conflict-free b64 LDS reads across 32 lanes

typedef __attribute__((ext_vector_type(2))) float v2f;
typedef __attribute__((ext_vector_type(8))) float v8f;

// ---------------- degree / dinv ----------------

__global__ void k_zero_deg(int* __restrict__ deg, int n) {
  int i = blockIdx.x * blockDim.x + threadIdx.x;
  if (i < n) deg[i] = 0;
}

__global__ void k_deg(const int* __restrict__ rows, const int* __restrict__ cols,
                      int* __restrict__ deg, int E) {
  int e = blockIdx.x * blockDim.x + threadIdx.x;
  if (e >= E) return;
  int r = rows[e], c = cols[e];
  if (r != c) atomicAdd(&deg[c], 1);
}

__global__ void k_dinv(const int* __restrict__ deg, float* __restrict__ dinv, int n) {
  int i = blockIdx.x * blockDim.x + threadIdx.x;
  if (i < n) dinv[i] = rsqrtf((float)deg[i] + 1.0f);
}

// ---------------- LoRA GEMM via V_WMMA_F32_16X16X4_F32 ----------------
// One wave32 per 16-node tile.
// W_B^T is staged zero-padded (16 cols) into LDS once per block so the inner
// loop is branch-free: global_load_b64 (x) + ds_load_b64 (W_B) + v_wmma.
// Stage 1: T(16x16) = X_tile(16x256) @ W_B^T(256x16pad), 64 WMMAs (K=4 steps).
// Stage 2: H(16x64) = T(16x4pad) @ W_A^T(4x64pad), 4 WMMAs.
// Epilogue: g = dinv * H -> ws (scatter source) and d_out (self-loop init).
__global__ void __launch_bounds__(256) k_lora_wmma(
    const float* __restrict__ x, const float* __restrict__ WB,
    const float* __restrict__ WA, const float* __restrict__ dinv,
    float* __restrict__ g, float* __restrict__ out, int ntiles) {
  __shared__ float ldsWB[16][WB_STRIDE];  // padded W_B (rows >= RANK are zero)
  __shared__ float ldsT[8][256];          // per-wave 16x16 T tile

  // Cooperative fill of padded W_B: 16 rows x 256 cols.
  for (int idx = threadIdx.x; idx < 16 * IN_DIM; idx += blockDim.x) {
    const int r = idx >> 8;          // 0..15
    const int k = idx & (IN_DIM - 1);
    ldsWB[r][k] = (r < RANK) ? WB[r * IN_DIM + k] : 0.0f;
  }
  __syncthreads();

  const int wave = threadIdx.x >> 5;
  const int lane = threadIdx.x & 31;
  int tile = blockIdx.x * 8 + wave;
  if (tile >= ntiles) tile = ntiles - 1;  // clamp: redundant recompute, EXEC stays full

  const int m    = lane & 15;   // A-layout: matrix row; B/C/D-layout: column
  const int half = lane >> 4;   // 0: K=0,1 / M=0..7 ; 1: K=2,3 / M=8..15
  const int koff = half * 2;
  const int base = tile * 16;

  // Stage 1: accumulate T = X @ WB^T over K=256 in steps of 4.
  const float* xrow = x + (size_t)(base + m) * IN_DIM + koff;  // 8B-aligned
  const float* wrow = &ldsWB[m][koff];                         // 8B-aligned

  v8f c = {};
  #pragma unroll 8
  for (int kk = 0; kk < IN_DIM; kk += 4) {
    v2f a = *reinterpret_cast<const v2f*>(xrow + kk);  // global_load_b64
    v2f b = *reinterpret_cast<const v2f*>(wrow + kk);  // ds_load_b64 (bank-conflict free)
    c = __builtin_amdgcn_wmma_f32_16x16x4_f32(false, a, false, b, (short)0, c,
                                              false, false);
  }

  // C-layout -> row-major LDS: VGPR j holds (M = j + 8*half, N = m).
  #pragma unroll
  for (int j = 0; j < 8; ++j) {
    ldsT[wave][(j + 8 * half) * 16 + m] = c[j];
  }
  __syncthreads();

  // Stage 2 A operand: A2[m][k] = T[m][k] (cols >= RANK are already zero).
  v2f a2;
  a2.x = ldsT[wave][m * 16 + koff];
  a2.y = ldsT[wave][m * 16 + koff + 1];

  float dv[8];
  #pragma unroll
  for (int j = 0; j < 8; ++j) dv[j] = dinv[base + j + 8 * half];

  #pragma unroll
  for (int jt = 0; jt < 4; ++jt) {
    const int oc = jt * 16 + m;  // output column for this lane
    v2f b2;
    b2.x = WA[oc * RANK + koff];                                 // koff in {0,2} < RANK
    b2.y = (koff + 1 < RANK) ? WA[oc * RANK + koff + 1] : 0.0f;  // k=3 pad -> 0
    v8f d = {};
    d = __builtin_amdgcn_wmma_f32_16x16x4_f32(false, a2, false, b2, (short)0, d,
                                              false, false);
    #pragma unroll
    for (int j = 0; j < 8; ++j) {
      const int node = base + j + 8 * half;
      const float val = d[j] * dv[j];
      const size_t off = (size_t)node * OUT_DIM + oc;
      g[off]   = val;  // gather source for the edge scatter
      out[off] = val;  // self-loop contribution initializes the accumulator
    }
  }
}

// Scalar fallback for N % 16 tail nodes (not used for N = 100000).
__global__ void k_lora_tail(const float* __restrict__ x, const float* __restrict__ WB,
                            const float* __restrict__ WA, const float* __restrict__ dinv,
                            float* __restrict__ g, float* __restrict__ out,
                            int start, int n) {
  int i = start + blockIdx.x * blockDim.x + threadIdx.x;
  if (i >= n) return;
  float t[RANK];
  for (int r = 0; r < RANK; ++r) {
    float s = 0.0f;
    for (int k = 0; k < IN_DIM; ++k) s += x[(size_t)i * IN_DIM + k] * WB[r * IN_DIM + k];
    t[r] = s;
  }
  const float dv = dinv[i];
  for (int oc = 0; oc < OUT_DIM; ++oc) {
    float s = 0.0f;
    for (int r = 0; r < RANK; ++r) s += t[r] * WA[oc * RANK + r];
    s *= dv;
    g[(size_t)i * OUT_DIM + oc] = s;
    out[(size_t)i * OUT_DIM + oc] = s;
  }
}

// ---------------- edge scatter: out[col] += g[row] ----------------
// 16 threads per edge, float4 per thread -> coalesced 256B gather per edge.
__global__ void k_scatter(const int* __restrict__ rows, const int* __restrict__ cols,
                          const float* __restrict__ g, float* __restrict__ out, int E) {
  int t = blockIdx.x * blockDim.x + threadIdx.x;
  int e = t >> 4;
  if (e >= E) return;
  int r = rows[e], c = cols[e];
  if (r == c) return;  // self loops dropped (re-added via init in k_lora)
  int sub = (t & 15) * 4;
  const float4 gv = *reinterpret_cast<const float4*>(g + (size_t)r * OUT_DIM + sub);
  float* o = out + (size_t)c * OUT_DIM + sub;
  atomicAdd(o + 0, gv.x);
  atomicAdd(o + 1, gv.y);
  atomicAdd(o + 2, gv.z);
  atomicAdd(o + 3, gv.w);
}

// ---------------- final scale + bias ----------------
__global__ void k_final(float* __restrict__ out, const float* __restrict__ dinv,
                        const float* __restrict__ bias, int total) {
  int t = blockIdx.x * blockDim.x + threadIdx.x;
  if (t >= total) return;
  out[t] = out[t] * dinv[t >> 6] + bias[t & (OUT_DIM - 1)];
}

// ---------------- host launcher ----------------
extern "C" void kernel_launch(void* const* d_in, const int* in_sizes, int n_in,
                              void* d_out, int out_size, void* d_ws, size_t ws_size,
                              hipStream_t stream) {
  const float* x    = (const float*)d_in[0];
  const int*   eidx = (const int*)d_in[1];
  const float* WB   = (const float*)d_in[2];
  const float* WA   = (const float*)d_in[3];
  const float* bias = (const float*)d_in[4];
  float* out = (float*)d_out;

  const int N = in_sizes[0] / IN_DIM;
  const int E = in_sizes[1] / 2;
  const int* rows = eidx;
  const int* cols = eidx + E;

  // workspace layout: g[N*64] | deg[N] | dinv[N]
  char* ws = (char*)d_ws;
  size_t gBytes = ((size_t)N * OUT_DIM * sizeof(float) + 255) & ~(size_t)255;
  size_t dBytes = ((size_t)N * sizeof(int) + 255) & ~(size_t)255;
  float* g    = (float*)ws;
  int*   deg  = (int*)(ws + gBytes);
  float* dinv = (float*)(ws + gBytes + dBytes);

  const int B = 256;
  k_zero_deg<<<(N + B - 1) / B, B, 0, stream>>>(deg, N);
  k_deg<<<(E + B - 1) / B, B, 0, stream>>>(rows, cols, deg, E);
  k_dinv<<<(N + B - 1) / B, B, 0, stream>>>(deg, dinv, N);

  const int ntiles = N / 16;
  if (ntiles > 0) {
    const int nblocks = (ntiles + 7) / 8;  // 8 waves per block, 1 tile per wave
    k_lora_wmma<<<nblocks, B, 0, stream>>>(x, WB, WA, dinv, g, out, ntiles);
  }
  const int rem_start = ntiles * 16;
  if (rem_start < N) {
    k_lora_tail<<<(N - rem_start + B - 1) / B, B, 0, stream>>>(x, WB, WA, dinv, g, out,
                                                               rem_start, N);
  }

  const long long sthreads = (long long)E * 16;
  k_scatter<<<(int)((sthreads + B - 1) / B), B, 0, stream>>>(rows, cols, g, out, E);

  const int total = N * OUT_DIM;
  k_final<<<(total + B - 1) / B, B, 0, stream>>>(out, dinv, bias, total);
}